// LiftLayer_31009663877640
// MI455X (gfx1250) — compile-verified
//
#include <hip/hip_runtime.h>

typedef __attribute__((ext_vector_type(2))) float v2f;
typedef __attribute__((ext_vector_type(8))) float v8f;

#define C_DIM 128
#define KSTEPS (C_DIM / 4)   // 32 WMMA k-steps

// ---------------------------------------------------------------------------
// Phase 1: per-node projections via V_WMMA_F32_16X16X4_F32 (fp32-exact).
//   s[n] = dot(x[n,:], att[0:128]);  t[n] = dot(x[n,:], att[128:256])
// One wave32 per 16-row tile. The B fragments (two weight vectors zero-padded
// to 16 columns) are identical for every tile: build all 32 of them, already
// masked, in LDS once per block. The hot loop is then exactly
//   global_load_b64 (A) + ds_load_b64 (B) + v_wmma per k-step.
// ---------------------------------------------------------------------------
__global__ __launch_bounds__(256) void node_proj_wmma(
    const float* __restrict__ x, const float* __restrict__ att,
    float* __restrict__ s, float* __restrict__ t,
    int n_nodes, int n_tiles)
{
    __shared__ v2f bfrag[KSTEPS][32];        // 8 KB, bank-conflict-free

    const int lane = threadIdx.x & 31;
    const int wave = threadIdx.x >> 5;

    // Cooperative build: 1024 entries / 256 threads = 4 each.
    // Entry (step, l): col = l&15, koff = 2*(l>>4);
    //   col 0 -> w_src = att[0:128], col 1 -> w_tgt = att[128:256], else 0.
    #pragma unroll
    for (int idx = threadIdx.x; idx < KSTEPS * 32; idx += 256) {
        const int step = idx >> 5;
        const int l    = idx & 31;
        const int col  = l & 15;
        const int ko   = (l >> 4) << 1;
        v2f v;
        if (col < 2) {
            v = *(const v2f*)(att + col * C_DIM + ko + 4 * step);
        } else {
            v.x = 0.0f; v.y = 0.0f;
        }
        bfrag[step][l] = v;
    }
    __syncthreads();                          // before any wave may exit

    const int tile = blockIdx.x * 8 + wave;
    if (tile >= n_tiles) return;              // wave-uniform: EXEC stays all-1s

    const int row0 = tile * 16;
    // A fragment addressing: lane L -> row (L&15), K-pair offset 2*(L>>4)
    int arow = row0 + (lane & 15);
    if (arow >= n_nodes) arow = n_nodes - 1;  // safe clamp (tail tiles)
    const int koff = (lane >> 4) << 1;        // 0 or 2

    const float* aptr = x + (size_t)arow * C_DIM + koff;

    v8f acc = {};
    #pragma unroll
    for (int i = 0; i < KSTEPS; ++i) {
        v2f a = *(const v2f*)(aptr + 4 * i);
        v2f b = bfrag[i][lane];
        // 8 args: (neg_a, A, neg_b, B, c_mod, C, reuse_a, reuse_b)
        acc = __builtin_amdgcn_wmma_f32_16x16x4_f32(
            false, a, false, b, (short)0, acc, false, false);
    }

    // C/D layout: element (M,N) lives in VGPR (M&7) of lane ((M>>3)*16 + N).
    //   lane 0  -> s[row0+0..7]   lane 16 -> s[row0+8..15]
    //   lane 1  -> t[row0+0..7]   lane 17 -> t[row0+8..15]
    float* obase = nullptr;
    int    roff  = 0;
    if      (lane == 0)  { obase = s; roff = row0;     }
    else if (lane == 16) { obase = s; roff = row0 + 8; }
    else if (lane == 1)  { obase = t; roff = row0;     }
    else if (lane == 17) { obase = t; roff = row0 + 8; }
    if (obase) {
        #pragma unroll
        for (int i = 0; i < 8; ++i) {
            const int r = roff + i;
            if (r < n_nodes) obase[r] = acc[i];
        }
    }
}

// ---------------------------------------------------------------------------
// Phase 2: out[e] = relu(s[src[e]] + t[tgt[e]]).  The 2 x 400 KB tables live
// in cache (192 MB L2); the streamed index reads (12.8 MB) and output writes
// (3.2 MB) get non-temporal hints so they don't evict the gather tables.
// ---------------------------------------------------------------------------
__global__ __launch_bounds__(256) void edge_gather(
    const long long* __restrict__ src, const long long* __restrict__ tgt,
    const float* __restrict__ s, const float* __restrict__ t,
    float* __restrict__ out, int n_edges)
{
    const int e = blockIdx.x * blockDim.x + threadIdx.x;
    if (e < n_edges) {
        const long long si = __builtin_nontemporal_load(src + e);
        const long long ti = __builtin_nontemporal_load(tgt + e);
        const float v = s[si] + t[ti];
        __builtin_nontemporal_store(v > 0.0f ? v : 0.0f, out + e);
    }
}

extern "C" void kernel_launch(void* const* d_in, const int* in_sizes, int n_in,
                              void* d_out, int out_size, void* d_ws, size_t ws_size,
                              hipStream_t stream)
{
    const float*     x   = (const float*)d_in[0];       // (N_NODES, 128) f32
    const long long* src = (const long long*)d_in[1];   // (N_EDGES,) i64
    const long long* tgt = (const long long*)d_in[2];   // (N_EDGES,) i64
    const float*     att = (const float*)d_in[3];       // (256, 1) f32

    float* out = (float*)d_out;

    const int n_nodes = in_sizes[0] / C_DIM;
    const int n_edges = in_sizes[1];

    float* s = (float*)d_ws;            // n_nodes floats
    float* t = s + n_nodes;             // n_nodes floats (400000 B -> 16B aligned)

    const int n_tiles = (n_nodes + 15) / 16;
    const int blocks1 = (n_tiles + 7) / 8;     // 8 waves (tiles) per 256-thr block
    node_proj_wmma<<<blocks1, 256, 0, stream>>>(x, att, s, t, n_nodes, n_tiles);

    const int blocks2 = (n_edges + 255) / 256;
    edge_gather<<<blocks2, 256, 0, stream>>>(src, tgt, s, t, out, n_edges);
}